// GNO2d_55619826483251
// MI455X (gfx1250) — compile-verified
//
#include <hip/hip_runtime.h>
#include <math.h>

// ---------------------------------------------------------------------------
// FNO spectral block as partial-DFT GEMMs on CDNA5 f32 WMMA (16x16x4).
//   P=8 B=4 C=32 H=W=256, M1=M2=32.  NI = P*B*C = 1024 images.
// Pipeline: fwd-W GEMM -> fwd-H complex GEMM -> per-mode channel mix (VALU)
//           -> inv-H complex GEMM -> inv-W (Hermitian real) GEMM.
// ---------------------------------------------------------------------------

typedef __attribute__((ext_vector_type(2))) float v2f;
typedef __attribute__((ext_vector_type(8))) float v8f;

#define PP 8
#define BB 4
#define CC 32
#define HH 256
#define WW 256
#define NIMG 1024
#define NROW (NIMG * HH)        // 262144 rows of the big W-direction GEMMs

// workspace layout (float offsets); total = 42,041,344 floats ~= 169 MB
#define OFF_FWT   0                          // [64 col][256 w]   fwd-W factors (transposed)
#define OFF_FHRE  (OFF_FWT  + 64*256)        // [64 ky][256 h]    fwd-H twiddle re
#define OFF_FHIM  (OFF_FHRE + 64*256)        // [64 ky][256 h]    fwd-H twiddle im
#define OFF_FHIRE (OFF_FHIM + 64*256)        // [256 h][64 ky]    inv-H twiddle re
#define OFF_FHIIM (OFF_FHIRE + 256*64)       // [256 h][64 ky]    inv-H twiddle im
#define OFF_FWIT  (OFF_FHIIM + 256*64)       // [256 w][64 k]     inv-W factors (transposed)
#define OFF_Y     (OFF_FWIT  + 256*64)       // [NROW][64]  (re0..31 | im0..31)
#define OFF_Z     (OFF_Y  + (size_t)NROW*64) // [NIMG][64 ky][64]
#define OFF_ZM    (OFF_Z  + (size_t)NIMG*64*64)
#define OFF_YI    (OFF_ZM + (size_t)NIMG*64*64) // [NROW][64]

__device__ __forceinline__ v8f wmma4(v2f a, v2f b, v8f c) {
  // D = A(16x4 f32) * B(4x16 f32) + C(16x16 f32)
  return __builtin_amdgcn_wmma_f32_16x16x4_f32(false, a, false, b, (short)0, c,
                                               false, false);
}

// ---------------------------------------------------------------------------
// Init: DFT factor matrices (run every call; deterministic)
// ---------------------------------------------------------------------------
__global__ void k_init_factors(float* __restrict__ ws) {
  const int idx = blockIdx.x * blockDim.x + threadIdx.x;
  const float TWO_PI = 6.28318530717958647692f;
  if (idx < 64 * 256) {
    // FwT[col][w]: rfft along W, modes kx<32.  e^{-i t}: re=cos, im=-sin
    const int col = idx >> 8, w = idx & 255;
    const int k = col & 31;
    float s, c;
    sincosf(TWO_PI * (float)((w * k) & 255) / 256.0f, &s, &c);
    ws[OFF_FWT + idx] = (col < 32) ? c : -s;
  } else if (idx < 3 * 64 * 256) {
    // Fh[kyIdx][h], kyIdx 0..31 -> ky, 32..63 -> ky+192.  e^{-i t}
    int j = idx - 64 * 256;
    const int im = j >= 64 * 256; j &= (64 * 256 - 1);
    const int kyIdx = j >> 8, h = j & 255;
    const int ky = (kyIdx < 32) ? kyIdx : (kyIdx + 192);
    float s, c;
    sincosf(TWO_PI * (float)((ky * h) & 255) / 256.0f, &s, &c);
    ws[(im ? OFF_FHIM : OFF_FHRE) + kyIdx * 256 + h] = im ? -s : c;
  } else if (idx < 3 * 64 * 256 + 2 * 256 * 64) {
    // Fhi[h][kyIdx]: inverse along H (full complex, 64 kept modes).  e^{+i t}
    int j = idx - 3 * 64 * 256;
    const int im = j >= 256 * 64; j &= (256 * 64 - 1);
    const int h = j >> 6, kyIdx = j & 63;
    const int ky = (kyIdx < 32) ? kyIdx : (kyIdx + 192);
    float s, c;
    sincosf(TWO_PI * (float)((ky * h) & 255) / 256.0f, &s, &c);
    ws[(im ? OFF_FHIIM : OFF_FHIRE) + j] = im ? s : c;
  } else {
    // FwiT[w][k]: irfft along W with Hermitian weights + 1/(H*W)
    const int j = idx - (3 * 64 * 256 + 2 * 256 * 64);
    const int w = j >> 6, k = j & 63;
    const int kx = k & 31;
    const float wgt = (kx == 0) ? 1.0f : 2.0f;
    const float inv = 1.0f / 65536.0f;  // 1/(H*W)
    float s, c;
    sincosf(TWO_PI * (float)((w * kx) & 255) / 256.0f, &s, &c);
    ws[OFF_FWIT + j] = (k < 32) ? wgt * inv * c : -wgt * inv * s;
  }
}

// ---------------------------------------------------------------------------
// Stage 1: Y[row][col] = sum_w x[row][w] * Fw[w][col]   (real GEMM, WMMA)
//   row = img*256 + h (262144), col = 64 (32 re | 32 im), K = 256
// ---------------------------------------------------------------------------
__global__ void k_fwd_w(const float* __restrict__ x, const float* __restrict__ ws,
                        float* __restrict__ Y) {
  __shared__ __align__(16) float sB[16 * 256];  // FwT rows col0..col0+15
  const int col0 = blockIdx.y * 16;
  for (int i = threadIdx.x; i < 16 * 256; i += 256)
    sB[i] = ws[OFF_FWT + col0 * 256 + i];
  __syncthreads();

  const int wave = threadIdx.x >> 5;
  const int lane = threadIdx.x & 31;
  const int l16 = lane & 15, half = lane >> 4, bk = half * 2;
  const int row0 = (blockIdx.x * 8 + wave) * 16;

  const float* arow = x + (size_t)(row0 + l16) * WW;
  v8f acc = {};
#pragma unroll 4
  for (int w0 = 0; w0 < WW; w0 += 4) {
    v2f a = *(const v2f*)(arow + w0 + bk);                 // A[m][w0+bk..+1]
    v2f b = *(const v2f*)(&sB[l16 * 256 + w0 + bk]);       // B[w0+bk..+1][n]
    acc = wmma4(a, b, acc);
  }
  const int col = col0 + l16;
  const int rb = row0 + half * 8;
#pragma unroll
  for (int r = 0; r < 8; ++r) Y[(size_t)(rb + r) * 64 + col] = acc[r];
}

// ---------------------------------------------------------------------------
// Stage 2: per image, Z[ky][kx] = sum_h Fh[ky][h] * Y[h][kx]  (complex GEMM)
// ---------------------------------------------------------------------------
__global__ void k_fwd_h(const float* __restrict__ Y, const float* __restrict__ fhre,
                        const float* __restrict__ fhim, float* __restrict__ Z) {
  const int wave = threadIdx.x >> 5;
  const int lane = threadIdx.x & 31;
  const int l16 = lane & 15, half = lane >> 4, bk = half * 2;
  const int tid = blockIdx.x * 8 + wave;            // 8192 tiles
  const int img = tid >> 3, sub = tid & 7;
  const int ky0 = (sub >> 1) * 16, col0 = (sub & 1) * 16;

  const float* yb = Y + (size_t)img * HH * 64;
  const float* are_row = fhre + (ky0 + l16) * 256;
  const float* aim_row = fhim + (ky0 + l16) * 256;
  v8f c_rr = {}, c_ii = {}, c_ri = {}, c_ir = {};
  for (int h0 = 0; h0 < HH; h0 += 4) {
    v2f are = *(const v2f*)(are_row + h0 + bk);
    v2f aim = *(const v2f*)(aim_row + h0 + bk);
    const float* y0 = yb + (h0 + bk) * 64;
    v2f bre, bim;
    bre.x = y0[col0 + l16];       bre.y = y0[64 + col0 + l16];
    bim.x = y0[32 + col0 + l16];  bim.y = y0[96 + col0 + l16];
    c_rr = wmma4(are, bre, c_rr);
    c_ii = wmma4(aim, bim, c_ii);
    c_ri = wmma4(are, bim, c_ri);
    c_ir = wmma4(aim, bre, c_ir);
  }
  float* zb = Z + (size_t)img * 4096;
  const int col = col0 + l16;
#pragma unroll
  for (int r = 0; r < 8; ++r) {
    const int ky = ky0 + r + half * 8;
    zb[ky * 64 + col]      = c_rr[r] - c_ii[r];   // Re
    zb[ky * 64 + 32 + col] = c_ri[r] + c_ir[r];   // Im
  }
}

// ---------------------------------------------------------------------------
// Stage 3: per (p, kyIdx, kx): out[b][o] = sum_i Z[b][i] * w[i][o]  (complex)
//   B=4 rows -> VALU with LDS-staged inputs (WMMA M-util would be 25%).
// ---------------------------------------------------------------------------
__global__ void k_mix(const float* __restrict__ Z,
                      const float* __restrict__ w1re, const float* __restrict__ w1im,
                      const float* __restrict__ w4re, const float* __restrict__ w4im,
                      float* __restrict__ Zm) {
  __shared__ float sIn[BB * CC * 64];  // [b][i][64] = 32 KB
  const int p = blockIdx.x >> 6;
  const int kyIdx = blockIdx.x & 63;
  const int high = kyIdx >= 32;
  const int y = high ? (kyIdx - 32) : kyIdx;
  const float* wre = high ? w4re : w1re;
  const float* wim = high ? w4im : w1im;

  for (int e = threadIdx.x; e < BB * CC * 64; e += 256) {
    const int b = e >> 11, rest = e & 2047;
    const int i = rest >> 6, col = rest & 63;
    sIn[e] = Z[(size_t)((p * 4 + b) * 32 + i) * 4096 + kyIdx * 64 + col];
  }
  __syncthreads();

  const int kx = threadIdx.x & 31;
  const int og = threadIdx.x >> 5;  // 0..7
  // w1/w4 shape (i,o,p,y,x): flat = (i*32+o)*8192 + p*1024 + y*32 + x
  const size_t wbase = (size_t)p * 1024 + y * 32 + kx;
  for (int oi = 0; oi < 4; ++oi) {
    const int o = og * 4 + oi;
    float ar[4] = {0, 0, 0, 0}, ai[4] = {0, 0, 0, 0};
    for (int i = 0; i < CC; ++i) {
      const size_t widx = wbase + (size_t)(i * 32 + o) * 8192;
      const float wr = wre[widx], wq = wim[widx];
#pragma unroll
      for (int b = 0; b < 4; ++b) {
        const float zr = sIn[b * 2048 + i * 64 + kx];
        const float zi = sIn[b * 2048 + i * 64 + 32 + kx];
        ar[b] += zr * wr - zi * wq;
        ai[b] += zr * wq + zi * wr;
      }
    }
#pragma unroll
    for (int b = 0; b < 4; ++b) {
      float* zmb = Zm + (size_t)((p * 4 + b) * 32 + o) * 4096 + kyIdx * 64;
      zmb[kx] = ar[b];
      zmb[32 + kx] = ai[b];
    }
  }
}

// ---------------------------------------------------------------------------
// Stage 4: per image, Yi[h][kx] = sum_ky Fhi[h][ky] * Zm[ky][kx]  (complex GEMM)
// ---------------------------------------------------------------------------
__global__ void k_inv_h(const float* __restrict__ Zm, const float* __restrict__ fire,
                        const float* __restrict__ fiim, float* __restrict__ Yi) {
  const int wave = threadIdx.x >> 5;
  const int lane = threadIdx.x & 31;
  const int l16 = lane & 15, half = lane >> 4, bk = half * 2;
  const int tid = blockIdx.x * 8 + wave;            // 32768 tiles
  const int img = tid >> 5, sub = tid & 31;
  const int h0 = (sub >> 1) * 16, col0 = (sub & 1) * 16;

  const float* zb = Zm + (size_t)img * 4096;
  const float* are_row = fire + (h0 + l16) * 64;
  const float* aim_row = fiim + (h0 + l16) * 64;
  v8f c_rr = {}, c_ii = {}, c_ri = {}, c_ir = {};
#pragma unroll 4
  for (int k0 = 0; k0 < 64; k0 += 4) {
    v2f are = *(const v2f*)(are_row + k0 + bk);
    v2f aim = *(const v2f*)(aim_row + k0 + bk);
    const float* z0 = zb + (k0 + bk) * 64;
    v2f bre, bim;
    bre.x = z0[col0 + l16];       bre.y = z0[64 + col0 + l16];
    bim.x = z0[32 + col0 + l16];  bim.y = z0[96 + col0 + l16];
    c_rr = wmma4(are, bre, c_rr);
    c_ii = wmma4(aim, bim, c_ii);
    c_ri = wmma4(are, bim, c_ri);
    c_ir = wmma4(aim, bre, c_ir);
  }
  float* yb = Yi + (size_t)img * HH * 64;
  const int col = col0 + l16;
#pragma unroll
  for (int r = 0; r < 8; ++r) {
    const int h = h0 + r + half * 8;
    yb[h * 64 + col]      = c_rr[r] - c_ii[r];
    yb[h * 64 + 32 + col] = c_ri[r] + c_ir[r];
  }
}

// ---------------------------------------------------------------------------
// Stage 5: out[row][w] = sum_k Yi[row][k] * Fwi[k][w]   (real GEMM, WMMA)
// ---------------------------------------------------------------------------
__global__ void k_inv_w(const float* __restrict__ Yi, const float* __restrict__ ws,
                        float* __restrict__ out) {
  __shared__ __align__(16) float sB[16 * 64];  // FwiT rows w0..w0+15
  const int w0 = blockIdx.y * 16;
  for (int i = threadIdx.x; i < 16 * 64; i += 256)
    sB[i] = ws[OFF_FWIT + w0 * 64 + i];
  __syncthreads();

  const int wave = threadIdx.x >> 5;
  const int lane = threadIdx.x & 31;
  const int l16 = lane & 15, half = lane >> 4, bk = half * 2;
  const int row0 = (blockIdx.x * 8 + wave) * 16;

  const float* arow = Yi + (size_t)(row0 + l16) * 64;
  v8f acc = {};
#pragma unroll 4
  for (int k0 = 0; k0 < 64; k0 += 4) {
    v2f a = *(const v2f*)(arow + k0 + bk);
    v2f b = *(const v2f*)(&sB[l16 * 64 + k0 + bk]);
    acc = wmma4(a, b, acc);
  }
  const int wcol = w0 + l16;
  const int rb = row0 + half * 8;
#pragma unroll
  for (int r = 0; r < 8; ++r) out[(size_t)(rb + r) * 256 + wcol] = acc[r];
}

// ---------------------------------------------------------------------------
extern "C" void kernel_launch(void* const* d_in, const int* in_sizes, int n_in,
                              void* d_out, int out_size, void* d_ws, size_t ws_size,
                              hipStream_t stream) {
  (void)in_sizes; (void)n_in; (void)out_size; (void)ws_size;
  const float* x    = (const float*)d_in[0];
  const float* w1re = (const float*)d_in[1];
  const float* w1im = (const float*)d_in[2];
  const float* w4re = (const float*)d_in[3];
  const float* w4im = (const float*)d_in[4];
  float* ws  = (float*)d_ws;   // requires ~169 MB of workspace
  float* out = (float*)d_out;

  k_init_factors<<<384, 256, 0, stream>>>(ws);
  k_fwd_w<<<dim3(2048, 4), 256, 0, stream>>>(x, ws, ws + OFF_Y);
  k_fwd_h<<<1024, 256, 0, stream>>>(ws + OFF_Y, ws + OFF_FHRE, ws + OFF_FHIM,
                                    ws + OFF_Z);
  k_mix<<<512, 256, 0, stream>>>(ws + OFF_Z, w1re, w1im, w4re, w4im, ws + OFF_ZM);
  k_inv_h<<<4096, 256, 0, stream>>>(ws + OFF_ZM, ws + OFF_FHIRE, ws + OFF_FHIIM,
                                    ws + OFF_YI);
  k_inv_w<<<dim3(2048, 16), 256, 0, stream>>>(ws + OFF_YI, ws, out);
}